// PCN_25168508354980
// MI455X (gfx1250) — compile-verified
//
#include <hip/hip_runtime.h>

typedef __attribute__((ext_vector_type(2))) float v2f;
typedef __attribute__((ext_vector_type(8))) float v8f;

// out[c] = sum_h clf_w[c,h] * states[9,h] + clf_b[c]
// The reference scan never touches states[9] (update mask excludes the last
// layer), so the whole 10-cycle recurrence is dead code. One wave32, three
// chained V_WMMA_F32_16X16X4_F32 (K = 4+4+2, zero padded), bias added at the
// store. Pure launch-latency bound; f32 WMMA keeps exact f32 math.
__global__ __launch_bounds__(32)
void PCN_25168508354980_kernel(const float* __restrict__ states,
                               const float* __restrict__ clf_w,
                               const float* __restrict__ clf_b,
                               float* __restrict__ out) {
    const int lane = threadIdx.x;   // 0..31, full wave
    const int m    = lane & 15;     // A-matrix row held by this lane
    const int half = lane >> 4;     // 0: lanes 0-15, 1: lanes 16-31

    const bool  mrow = (m < 10);
    const float* arow = clf_w + (mrow ? m : 0) * 10;  // clamped: no OOB reads
    const float* s9   = states + 9 * 10;              // top-layer state

    v8f c = {};  // bias applied in the epilogue

    // A 16x4: lane (m,half), vgpr i -> A[m][2*half + i]
    // B 4x16: row K in vgpr K&1, lane-half K>>1 -> b[i] = s9[2*half + i] (bcast over N)
    // K only overflows (k=10,11) in chunk kk==2 on the upper lane-half.
#pragma unroll
    for (int kk = 0; kk < 3; ++kk) {
        const bool kin = (kk < 2) || (half == 0);
        v2f a, b;
#pragma unroll
        for (int i = 0; i < 2; ++i) {
            const int k     = 4 * kk + 2 * half + i;
            const int ksafe = kin ? k : 0;
            const float av  = arow[ksafe];
            const float bv  = s9[ksafe];
            a[i] = (kin && mrow) ? av : 0.0f;
            b[i] = kin ? bv : 0.0f;
        }
        // 8 args: (neg_a, A, neg_b, B, c_mod, C, reuse_a, reuse_b)
        c = __builtin_amdgcn_wmma_f32_16x16x4_f32(false, a, false, b,
                                                  (short)0, c, false, false);
    }

    // Every column of D is identical (B broadcast over N); read column N=0.
    // Lane 0 holds rows 0..7 in c[0..7]; lane 16 holds rows 8..9 in c[0..1].
    if (lane == 0) {
        const v8f bias = *(const v8f*)clf_b;   // clf_b[0..7], aligned (hipMalloc)
        *(v8f*)out = c + bias;                 // out[0..7]: 2x global_store_b128
    } else if (lane == 16) {
        const v2f bias = *(const v2f*)(clf_b + 8);
        v2f r;
        r[0] = c[0] + bias[0];
        r[1] = c[1] + bias[1];
        *(v2f*)(out + 8) = r;                  // out[8..9]: global_store_b64
    }
}

extern "C" void kernel_launch(void* const* d_in, const int* in_sizes, int n_in,
                              void* d_out, int out_size, void* d_ws, size_t ws_size,
                              hipStream_t stream) {
    // setup_inputs() order: 0=input_data (unused), 1=states, 2=W (unused: the
    // scan never affects states[9]), 3=clf_w, 4=clf_b.
    const float* states = (const float*)d_in[1];
    const float* clf_w  = (const float*)d_in[3];
    const float* clf_b  = (const float*)d_in[4];
    float*       out    = (float*)d_out;

    PCN_25168508354980_kernel<<<1, 32, 0, stream>>>(states, clf_w, clf_b, out);
}